// LengthRegulator_45672682226289
// MI455X (gfx1250) — compile-verified
//
#include <hip/hip_runtime.h>
#include <hip/hip_bf16.h>

typedef __attribute__((ext_vector_type(2))) float v2f;
typedef __attribute__((ext_vector_type(8))) float v8f;

#define B_ 16
#define L_ 512
#define D_ 256
#define F_ 256
#define T_ 4096
#define KTOT 768           // 3 taps * 256 channels
#define LN_EPS 1e-5f

// ---------------------------------------------------------------------------
// Kernel 1: conv1d(K=3, SAME) + bias + ReLU + LayerNorm  -> h1 [B,L,F]
// grid = B * (L/16) = 512 blocks, 512 threads (16 waves). Each block computes
// a 16(M) x 256(N) output tile; wave w owns N columns [16w, 16w+16).
// ---------------------------------------------------------------------------
__launch_bounds__(512)
__global__ void lr_conv_relu_ln1(const float* __restrict__ x,    // [B,L,D]
                                 const float* __restrict__ w,    // [3,D,F] == [768,256]
                                 const float* __restrict__ bias, // [F]
                                 const float* __restrict__ g,
                                 const float* __restrict__ beta,
                                 float* __restrict__ h)          // [B,L,F]
{
    __shared__ float xr[18 * 256];   // rows l0-1 .. l0+16 (zero-padded halo)
    __shared__ float hbuf[16 * 256]; // post-ReLU tile for LayerNorm

    const int blk  = blockIdx.x;
    const int b    = blk >> 5;            // 32 blocks per batch
    const int l0   = (blk & 31) << 4;     // M-tile base row
    const int tid  = threadIdx.x;
    const int wave = tid >> 5;            // 0..15
    const int lane = tid & 31;

    // Stage 18 rows of x (with SAME zero padding) into LDS.
    for (int i = tid; i < 18 * 256; i += 512) {
        const int row = i >> 8;           // 0..17 -> l = l0 + row - 1
        const int d   = i & 255;
        const int l   = l0 + row - 1;
        xr[i] = (l >= 0 && l < L_) ? x[((size_t)b * L_ + l) * D_ + d] : 0.0f;
    }
    __syncthreads();

    const int half    = lane >> 4;        // 0: K pair {0,1}, 1: K pair {2,3}
    const int lid     = lane & 15;        // M row (for A) / N col (for B)
    const int colBase = wave << 4;

    v8f c = {};
    for (int kk = 0; kk < KTOT; kk += 4) {
        const int k0  = kk + (half << 1);
        const int tap = k0 >> 8;          // 0..2
        const int d0  = k0 & 255;
        // A[m=lid][k0], A[m=lid][k0+1] : output row lid uses input row lid+tap-1+1
        const float* ar = &xr[(lid + tap) * 256 + d0];
        v2f a; a.x = ar[0]; a.y = ar[1];
        // B[k0][n], B[k0+1][n] : weights are already [K*D, F] row-major
        const float* br = &w[(size_t)k0 * F_ + colBase + lid];
        v2f bb; bb.x = br[0]; bb.y = br[F_];
        c = __builtin_amdgcn_wmma_f32_16x16x4_f32(false, a, false, bb,
                                                  (short)0, c, false, false);
    }

    // bias + ReLU -> LDS (C layout: vgpr r, lanes<16: M=r, lanes>=16: M=r+8; N=lane&15)
    {
        const int n  = colBase + lid;
        const float bn = bias[n];
        #pragma unroll
        for (int r = 0; r < 8; ++r) {
            const int m = r + (half << 3);
            float v = c[r] + bn;
            hbuf[m * 256 + n] = v > 0.0f ? v : 0.0f;
        }
    }
    __syncthreads();

    // LayerNorm over F: wave m normalizes row m (conflict-free strided reads).
    {
        const int m = wave;
        float s = 0.0f, sq = 0.0f;
        float vals[8];
        #pragma unroll
        for (int i = 0; i < 8; ++i) {
            const float v = hbuf[m * 256 + i * 32 + lane];
            vals[i] = v; s += v; sq += v * v;
        }
        #pragma unroll
        for (int off = 16; off > 0; off >>= 1) {
            s  += __shfl_xor(s,  off, 32);
            sq += __shfl_xor(sq, off, 32);
        }
        const float mean = s * (1.0f / 256.0f);
        const float var  = sq * (1.0f / 256.0f) - mean * mean;
        const float rstd = rsqrtf(var + LN_EPS);
        float* out = &h[((size_t)b * L_ + (l0 + m)) * F_];
        #pragma unroll
        for (int i = 0; i < 8; ++i) {
            const int d = i * 32 + lane;
            out[d] = (vals[i] - mean) * rstd * g[d] + beta[d];
        }
    }
}

// ---------------------------------------------------------------------------
// Kernel 2: conv1d(K=3,SAME) on h1 + bias + ReLU + LayerNorm + (F->1 linear)
// Writes only pred_duration [B,L].
// ---------------------------------------------------------------------------
__launch_bounds__(512)
__global__ void lr_conv_relu_ln2_lin(const float* __restrict__ h1,  // [B,L,F]
                                     const float* __restrict__ w,   // [768,256]
                                     const float* __restrict__ bias,
                                     const float* __restrict__ g,
                                     const float* __restrict__ beta,
                                     const float* __restrict__ lw,  // [F]
                                     const float* __restrict__ lb,  // [1]
                                     float* __restrict__ pred)      // [B,L]
{
    __shared__ float xr[18 * 256];
    __shared__ float hbuf[16 * 256];

    const int blk  = blockIdx.x;
    const int b    = blk >> 5;
    const int l0   = (blk & 31) << 4;
    const int tid  = threadIdx.x;
    const int wave = tid >> 5;
    const int lane = tid & 31;

    for (int i = tid; i < 18 * 256; i += 512) {
        const int row = i >> 8;
        const int d   = i & 255;
        const int l   = l0 + row - 1;
        xr[i] = (l >= 0 && l < L_) ? h1[((size_t)b * L_ + l) * F_ + d] : 0.0f;
    }
    __syncthreads();

    const int half    = lane >> 4;
    const int lid     = lane & 15;
    const int colBase = wave << 4;

    v8f c = {};
    for (int kk = 0; kk < KTOT; kk += 4) {
        const int k0  = kk + (half << 1);
        const int tap = k0 >> 8;
        const int d0  = k0 & 255;
        const float* ar = &xr[(lid + tap) * 256 + d0];
        v2f a; a.x = ar[0]; a.y = ar[1];
        const float* br = &w[(size_t)k0 * F_ + colBase + lid];
        v2f bb; bb.x = br[0]; bb.y = br[F_];
        c = __builtin_amdgcn_wmma_f32_16x16x4_f32(false, a, false, bb,
                                                  (short)0, c, false, false);
    }

    {
        const int n  = colBase + lid;
        const float bn = bias[n];
        #pragma unroll
        for (int r = 0; r < 8; ++r) {
            const int m = r + (half << 3);
            float v = c[r] + bn;
            hbuf[m * 256 + n] = v > 0.0f ? v : 0.0f;
        }
    }
    __syncthreads();

    // LayerNorm + fused F->1 linear per row.
    {
        const int m = wave;
        float s = 0.0f, sq = 0.0f;
        float vals[8];
        #pragma unroll
        for (int i = 0; i < 8; ++i) {
            const float v = hbuf[m * 256 + i * 32 + lane];
            vals[i] = v; s += v; sq += v * v;
        }
        #pragma unroll
        for (int off = 16; off > 0; off >>= 1) {
            s  += __shfl_xor(s,  off, 32);
            sq += __shfl_xor(sq, off, 32);
        }
        const float mean = s * (1.0f / 256.0f);
        const float var  = sq * (1.0f / 256.0f) - mean * mean;
        const float rstd = rsqrtf(var + LN_EPS);
        float acc = 0.0f;
        #pragma unroll
        for (int i = 0; i < 8; ++i) {
            const int d = i * 32 + lane;
            const float y = (vals[i] - mean) * rstd * g[d] + beta[d];
            acc += y * lw[d];
        }
        #pragma unroll
        for (int off = 16; off > 0; off >>= 1)
            acc += __shfl_xor(acc, off, 32);
        if (lane == 0)
            pred[b * L_ + l0 + m] = acc + lb[0];
    }
}

// ---------------------------------------------------------------------------
// Kernel 3: inclusive cumsum of target_duration per batch. grid = B, 512 thr.
// ---------------------------------------------------------------------------
__launch_bounds__(512)
__global__ void lr_cumsum(const int* __restrict__ dur, int* __restrict__ cum)
{
    __shared__ int buf[L_];
    const int b = blockIdx.x, tid = threadIdx.x;
    buf[tid] = dur[b * L_ + tid];
    __syncthreads();
    for (int off = 1; off < L_; off <<= 1) {
        const int v = (tid >= off) ? buf[tid - off] : 0;
        __syncthreads();
        buf[tid] += v;
        __syncthreads();
    }
    cum[b * L_ + tid] = buf[tid];
}

// ---------------------------------------------------------------------------
// Kernel 4: length regulation. One wave per (b, t) output row; binary-search
// the token index, then a coalesced 1KB float4 row copy (or zero fill).
// grid = B*T/8 = 8192 blocks of 256 threads.
// ---------------------------------------------------------------------------
__launch_bounds__(256)
__global__ void lr_regulate(const float* __restrict__ x,   // [B,L,D]
                            const int* __restrict__ cum,   // [B,L]
                            float* __restrict__ out)       // [B,T,D]
{
    const int gw   = blockIdx.x * 8 + (threadIdx.x >> 5); // 0 .. B*T-1
    const int lane = threadIdx.x & 31;
    const int b = gw >> 12;          // T = 4096
    const int t = gw & (T_ - 1);

    const int* c = &cum[b * L_];
    const int total = c[L_ - 1];

    float4* dst = (float4*)&out[(((size_t)b * T_) + t) * D_];
    if (t >= total) {
        const float4 z = make_float4(0.0f, 0.0f, 0.0f, 0.0f);
        dst[lane]      = z;
        dst[lane + 32] = z;
        return;
    }
    // searchsorted(cum, t, side='right'): first idx with cum[idx] > t
    int lo = 0, hi = L_;
    while (lo < hi) {
        const int mid = (lo + hi) >> 1;
        if (c[mid] <= t) lo = mid + 1; else hi = mid;
    }
    const int idx = lo > (L_ - 1) ? (L_ - 1) : lo;
    const float4* src = (const float4*)&x[((size_t)b * L_ + idx) * D_];
    dst[lane]      = src[lane];
    dst[lane + 32] = src[lane + 32];
}

// ---------------------------------------------------------------------------
extern "C" void kernel_launch(void* const* d_in, const int* in_sizes, int n_in,
                              void* d_out, int out_size, void* d_ws, size_t ws_size,
                              hipStream_t stream)
{
    const float* x    = (const float*)d_in[0];
    const int*   dur  = (const int*)  d_in[1];
    const float* c1w  = (const float*)d_in[2];
    const float* c1b  = (const float*)d_in[3];
    const float* g1   = (const float*)d_in[4];
    const float* b1   = (const float*)d_in[5];
    const float* c2w  = (const float*)d_in[6];
    const float* c2b  = (const float*)d_in[7];
    const float* g2   = (const float*)d_in[8];
    const float* b2   = (const float*)d_in[9];
    const float* lw   = (const float*)d_in[10];
    const float* lb   = (const float*)d_in[11];
    // d_in[12] = mel_max_len (fixed 4096 here)

    float* out  = (float*)d_out;
    float* reg  = out;                                    // [16,4096,256]
    float* pred = out + (size_t)B_ * T_ * D_;             // [16,512]

    float* h1  = (float*)d_ws;                            // 8 MB
    int*   cum = (int*)((char*)d_ws + (size_t)B_ * L_ * F_ * sizeof(float));

    lr_conv_relu_ln1    <<<B_ * (L_ / 16), 512, 0, stream>>>(x, c1w, c1b, g1, b1, h1);
    lr_conv_relu_ln2_lin<<<B_ * (L_ / 16), 512, 0, stream>>>(h1, c2w, c2b, g2, b2, lw, lb, pred);
    lr_cumsum           <<<B_, L_, 0, stream>>>(dur, cum);
    lr_regulate         <<<(B_ * T_) / 8, 256, 0, stream>>>(x, cum, reg);
}